// AAAProtectedClassifier_62646392979710
// MI455X (gfx1250) — compile-verified
//
#include <hip/hip_runtime.h>
#include <hip/hip_bf16.h>
#include <math.h>

// ---------------------------------------------------------------------------
// MI455X (gfx1250) implementation.
//  Kernel 1: z = x @ W + b via V_WMMA_F32_16X16X4_F32 (fp32 tensor path),
//            one wave32 per 16x16 output tile.
//  Kernel 2: per-row AAA loop. One wave32 per row; u/m/v/e fully register-
//            resident (32 elems/lane); all row reductions via __shfl_xor.
//            100 Adam steps with zero memory traffic between steps.
// ---------------------------------------------------------------------------

#define NROW 2048
#define KDIM 2048
#define NCOL 1000
#define NPAD 1024  // 32 lanes * 32 elems

#define F_ALPHA 1.0f
#define F_TAU   6.0f
#define F_BETA  5.0f
#define F_LR    0.1f
#define F_B1    0.9f
#define F_B2    0.999f
#define F_EPS   1e-8f
#define F_PI    3.14159265358979323846f

typedef float v2f __attribute__((ext_vector_type(2)));
typedef float v8f __attribute__((ext_vector_type(8)));

// ------------------------------ GEMM (WMMA) --------------------------------
// grid = (63, 128), block = 32 (one wave). Tile (row0..row0+15, col0..col0+15).
__global__ __launch_bounds__(32) void aaa_gemm_wmma(
    const float* __restrict__ x, const float* __restrict__ W,
    const float* __restrict__ b, float* __restrict__ z) {
  const int lane = threadIdx.x & 31;
  const int lm = lane & 15;      // M (A) / N (B,C) position within half-wave
  const int lh = lane >> 4;      // half-wave select
  const int col0 = blockIdx.x * 16;
  const int row0 = blockIdx.y * 16;

  // Clamp B column so padding tiles read in-bounds garbage (stores are masked).
  int cw = col0 + lm;
  if (cw > NCOL - 1) cw = NCOL - 1;

  const float* __restrict__ xrow = x + (long)(row0 + lm) * KDIM;

  v8f acc = {};
#pragma unroll 4
  for (int k0 = 0; k0 < KDIM; k0 += 4) {
    const int ka = k0 + 2 * lh;
    // A fragment: 16x4 fp32 -> 2 VGPRs/lane. VGPR0: K=ka, VGPR1: K=ka+1.
    v2f a;
    a.x = xrow[ka];
    a.y = xrow[ka + 1];
    // B fragment: 4x16 fp32 -> 2 VGPRs/lane (rows striped across lanes).
    v2f bf;
    bf.x = W[(long)ka * NCOL + cw];
    bf.y = W[(long)(ka + 1) * NCOL + cw];
    // D = A*B + C : v_wmma_f32_16x16x4_f32
    acc = __builtin_amdgcn_wmma_f32_16x16x4_f32(
        /*neg_a=*/false, a, /*neg_b=*/false, bf,
        /*c_mod=*/(short)0, acc, /*reuse_a=*/false, /*reuse_b=*/false);
  }

  // C/D layout: VGPR i -> M = i (lanes 0-15) / i+8 (lanes 16-31), N = lm.
  const int c = col0 + lm;
#pragma unroll
  for (int i = 0; i < 8; ++i) {
    const int r = row0 + i + 8 * lh;
    if (c < NCOL) z[(long)r * NCOL + c] = acc[i] + b[c];
  }
}

// --------------------------- per-row Adam loop -----------------------------
// grid = 256 blocks * 8 waves = 2048 rows. Wave per row, 32 elems per lane.
__global__ __launch_bounds__(256) void aaa_adam_rows(float* __restrict__ z) {
  const int lane = threadIdx.x & 31;
  const int row = blockIdx.x * 8 + (threadIdx.x >> 5);
  const long base = (long)row * NCOL;

  float u[32], m[32], v[32], e[32];
#pragma unroll
  for (int j = 0; j < 32; ++j) {
    const int c = j * 32 + lane;
    u[j] = (c < NCOL) ? z[base + c] : -INFINITY;
    m[j] = 0.0f;
    v[j] = 0.0f;
  }

  // ---- initial stats from z: y = argmax, l_org, l_trg, p_trg ----
  float lmax = -INFINITY;
  int lidx = NPAD;
#pragma unroll
  for (int j = 0; j < 32; ++j) {
    const int c = j * 32 + lane;
    if (u[j] > lmax) { lmax = u[j]; lidx = c; }
  }
  float rmax = lmax;
  int y = lidx;
#pragma unroll
  for (int off = 16; off > 0; off >>= 1) {
    const float ov = __shfl_xor(rmax, off, 32);
    const int oi = __shfl_xor(y, off, 32);
    if (ov > rmax || (ov == rmax && oi < y)) { rmax = ov; y = oi; }
  }
  float lnot = -INFINITY, lsum = 0.0f;
#pragma unroll
  for (int j = 0; j < 32; ++j) {
    const int c = j * 32 + lane;
    lsum += expf(u[j] - rmax);              // exp(-inf)=0 for padding
    if (c != y && u[j] > lnot) lnot = u[j];
  }
#pragma unroll
  for (int off = 16; off > 0; off >>= 1) {
    lnot = fmaxf(lnot, __shfl_xor(lnot, off, 32));
    lsum += __shfl_xor(lsum, off, 32);
  }
  const float l_org = rmax - lnot;
  const float l_atr = (floorf(l_org / F_TAU) + 0.5f) * F_TAU;
  const float l_trg =
      l_org - F_ALPHA * F_TAU * sinf(F_PI * (1.0f - 2.0f * (l_org - l_atr) / F_TAU));
  const float p_trg = 1.0f / lsum;          // TEMP = 1: pmax = exp(0)/sumexp

  // ---- 100 Adam steps, fully register resident ----
  float b1p = 1.0f, b2p = 1.0f;
#pragma unroll 1
  for (int t = 0; t < 100; ++t) {
    // pass 1: row max + argmax k, u[y], max excluding y
    float lm_ = -INFINITY, luy = -INFINITY, lny = -INFINITY;
    int li_ = NPAD;
#pragma unroll
    for (int j = 0; j < 32; ++j) {
      const int c = j * 32 + lane;
      if (u[j] > lm_) { lm_ = u[j]; li_ = c; }
      if (c == y) luy = u[j];
      else if (u[j] > lny) lny = u[j];
    }
    float rmx = lm_;
    int k = li_;
#pragma unroll
    for (int off = 16; off > 0; off >>= 1) {
      const float ov = __shfl_xor(rmx, off, 32);
      const int oi = __shfl_xor(k, off, 32);
      if (ov > rmx || (ov == rmx && oi < k)) { rmx = ov; k = oi; }
      luy = fmaxf(luy, __shfl_xor(luy, off, 32));
      lny = fmaxf(lny, __shfl_xor(lny, off, 32));
    }

    // pass 2: stable softmax denominator
    float ls = 0.0f;
#pragma unroll
    for (int j = 0; j < 32; ++j) {
      e[j] = expf(u[j] - rmx);
      ls += e[j];
    }
#pragma unroll
    for (int off = 16; off > 0; off >>= 1) ls += __shfl_xor(ls, off, 32);
    const float inv = 1.0f / ls;
    const float pmax = inv;                 // exp(rmx - rmx) / ls
    const float margin = luy - lny;

    const float dm = margin - l_trg;
    const float s1 = (dm > 0.0f) ? 1.0f : ((dm < 0.0f) ? -1.0f : 0.0f);
    const float dp = pmax - p_trg;
    const float s2 = F_BETA * ((dp > 0.0f) ? 1.0f : ((dp < 0.0f) ? -1.0f : 0.0f));

    b1p *= F_B1;
    b2p *= F_B2;
    const float c1 = 1.0f / (1.0f - b1p);
    const float c2 = 1.0f / (1.0f - b2p);

    // pass 3: gradient + Adam update (elementwise, registers only)
#pragma unroll
    for (int j = 0; j < 32; ++j) {
      const int c = j * 32 + lane;
      const float p = e[j] * inv;
      float g = s2 * pmax * (((c == k) ? 1.0f : 0.0f) - p);
      if (c == y) g += s1;
      m[j] = F_B1 * m[j] + (1.0f - F_B1) * g;
      v[j] = F_B2 * v[j] + (1.0f - F_B2) * g * g;
      const float mh = m[j] * c1;
      const float vh = v[j] * c2;
      u[j] -= F_LR * mh / (sqrtf(vh) + F_EPS);
    }
  }

#pragma unroll
  for (int j = 0; j < 32; ++j) {
    const int c = j * 32 + lane;
    if (c < NCOL) z[base + c] = u[j];
  }
}

// ------------------------------- launcher ----------------------------------
extern "C" void kernel_launch(void* const* d_in, const int* in_sizes, int n_in,
                              void* d_out, int out_size, void* d_ws, size_t ws_size,
                              hipStream_t stream) {
  (void)in_sizes; (void)n_in; (void)out_size; (void)d_ws; (void)ws_size;
  const float* x = (const float*)d_in[0];
  const float* W = (const float*)d_in[1];
  const float* b = (const float*)d_in[2];
  float* out = (float*)d_out;

  // 63 column tiles (1000 -> 63*16), 128 row tiles, one wave per tile.
  dim3 ggrid((NCOL + 15) / 16, NROW / 16);
  aaa_gemm_wmma<<<ggrid, 32, 0, stream>>>(x, W, b, out);

  // 2048 rows, 8 waves (rows) per 256-thread block.
  aaa_adam_rows<<<NROW / 8, 256, 0, stream>>>(out);
}